// MACUnit_33492154974825
// MI455X (gfx1250) — compile-verified
//
#include <hip/hip_runtime.h>
#include <hip/hip_bf16.h>

// MAC cell forward, restructured:
//   - read-unit 825-GFLOP GEMM algebraically collapsed to batched matvecs
//   - dense projections on V_WMMA_F32_16X16X4_F32 (exact fp32 semantics)
//     * templated weight layout (no runtime branches in K-loop)
//     * concat handled as two sequential K-phases
//     * 4 independent accumulator chains, 8 loads batched per 16-K block
//   - attention + softmax via LDS + wave32 shuffle reductions
//
// D=1024, NS=12, B=32, S=128, K=512

typedef __attribute__((ext_vector_type(2))) float v2f;
typedef __attribute__((ext_vector_type(8))) float v8f;

#define WMMA_F32_4(a, b, c)                                              \
  __builtin_amdgcn_wmma_f32_16x16x4_f32(false, (a), false, (b),          \
                                        (short)0, (c), false, false)

// ---------------------------------------------------------------------------
// One K-phase of the 16x16 tile GEMM.
//   ap : A row pointer, pre-offset by  m*lda + 2*half   (advances 16/iter)
//   wp : WT ? W + n*ldw + 2*half  (advances 16/iter)
//        else W + (2*half)*ldw + n (advances 16*ldw/iter)
// Kp must be a multiple of 16. 4 independent WMMA chains.
// ---------------------------------------------------------------------------
template <bool WT>
__device__ __forceinline__ void gemm_phase(
    const float* __restrict__ ap, const float* __restrict__ wp, int ldw,
    int Kp, v8f& acc0, v8f& acc1, v8f& acc2, v8f& acc3)
{
  for (int k = 0; k < Kp; k += 16) {
    v2f a[4], b[4];
#pragma unroll
    for (int j = 0; j < 4; ++j) {
      a[j].x = ap[4 * j + 0];
      a[j].y = ap[4 * j + 1];
      if (WT) {
        b[j].x = wp[4 * j + 0];
        b[j].y = wp[4 * j + 1];
      } else {
        b[j].x = wp[(size_t)(4 * j + 0) * ldw];
        b[j].y = wp[(size_t)(4 * j + 1) * ldw];
      }
    }
    acc0 = WMMA_F32_4(a[0], b[0], acc0);
    acc1 = WMMA_F32_4(a[1], b[1], acc1);
    acc2 = WMMA_F32_4(a[2], b[2], acc2);
    acc3 = WMMA_F32_4(a[3], b[3], acc3);
    ap += 16;
    wp += WT ? (size_t)16 : (size_t)16 * ldw;
  }
}

// ---------------------------------------------------------------------------
// C[m,n] = sum_k A(m,k)*B(k,n) + bias[n],  A = rowwise concat [A0 | A1]
//   B(k,n) = WT ? W[n*ldw + k] : W[k*ldw + n]
// One wave per 16x16 C tile; grid = (N/16, M/16). K0,K1 multiples of 16.
// outmode 1: n encodes (s,d) = (n>>10, n&1023); store C[s][m][d] (q_pa_all).
// ---------------------------------------------------------------------------
template <bool WT>
__global__ __launch_bounds__(32) void wmma_gemm_f32(
    const float* __restrict__ A0, int K0,
    const float* __restrict__ A1, int K1,
    const float* __restrict__ W, int ldw,
    const float* __restrict__ bias,
    float* __restrict__ C, int ldc, int outmode)
{
  const int n0 = blockIdx.x << 4;
  const int m0 = blockIdx.y << 4;
  const int lane = threadIdx.x;
  const int half = lane >> 4;   // which K-pair this lane supplies
  const int l = lane & 15;
  const int m = m0 + l;         // A row for this lane
  const int n = n0 + l;         // B col for this lane

  v8f acc0 = {}, acc1 = {}, acc2 = {}, acc3 = {};
  {
    const float* ap = A0 + (size_t)m * K0 + 2 * half;
    const float* wp = WT ? W + (size_t)n * ldw + 2 * half
                         : W + (size_t)(2 * half) * ldw + n;
    gemm_phase<WT>(ap, wp, ldw, K0, acc0, acc1, acc2, acc3);
  }
  if (K1 > 0) {                 // uniform scalar branch
    const float* ap = A1 + (size_t)m * K1 + 2 * half;
    const float* wp = WT ? W + (size_t)n * ldw + K0 + 2 * half
                         : W + (size_t)(K0 + 2 * half) * ldw + n;
    gemm_phase<WT>(ap, wp, ldw, K1, acc0, acc1, acc2, acc3);
  }
  const v8f acc = (acc0 + acc1) + (acc2 + acc3);

  const float bv = bias ? bias[n] : 0.0f;
#pragma unroll
  for (int r = 0; r < 8; ++r) {
    const int mr = m0 + r + 8 * half;   // C layout: VGPR r -> rows r / r+8
    const float val = acc[r] + bv;
    if (outmode == 0) {
      C[(size_t)mr * ldc + n] = val;
    } else {  // q_pa_all: [NS, B, D] with n = s*1024 + d
      const int s = n >> 10, d = n & 1023;
      C[(size_t)s * 32768 + (size_t)mr * 1024 + d] = val;
    }
  }
}

// ---------------------------------------------------------------------------
// control0 = broadcast(control_init); memory0 = q_pa_all[0]
// ---------------------------------------------------------------------------
__global__ __launch_bounds__(256) void init_state_kernel(
    const float* __restrict__ control_init,
    const float* __restrict__ qpa,
    float* __restrict__ control, float* __restrict__ mem0)
{
  const int i = blockIdx.x * 256 + threadIdx.x;   // 32768 total
  control[i] = control_init[0];
  mem0[i]    = qpa[i];
}

// ---------------------------------------------------------------------------
// Control unit (one block per batch element b):
//   u[d]      = cq[b,d] * ca_w[d]
//   att[s]    = softmax_s( ctx[b,s,:] . u + ca_b )
//   control'  = att . ctx[b]            -> control (in place)
//   v[b,d]    = control'[b,d] * ra_w[d]
//   h[b]      = v[b] . ri_b + ra_b      (read-unit affine constant)
// ---------------------------------------------------------------------------
__global__ __launch_bounds__(256) void control_unit_kernel(
    const float* __restrict__ cq, const float* __restrict__ context,
    const float* __restrict__ ca_w, const float* __restrict__ ca_b,
    const float* __restrict__ ra_w, const float* __restrict__ ri_b,
    const float* __restrict__ ra_b,
    float* __restrict__ control, float* __restrict__ v, float* __restrict__ h)
{
  __shared__ float u[1024];
  __shared__ float att[128];
  __shared__ float red[8];
  const int b = blockIdx.x, tid = threadIdx.x;
  const int lane = tid & 31, wid = tid >> 5;
  const float* ctx = context + (size_t)b * 128 * 1024;

  for (int d = tid; d < 1024; d += 256) u[d] = cq[b * 1024 + d] * ca_w[d];
  __syncthreads();

  for (int s = wid; s < 128; s += 8) {           // one wave per logit row
    float acc = 0.f;
    const float* row = ctx + (size_t)s * 1024;
    for (int d = lane; d < 1024; d += 32) acc = fmaf(u[d], row[d], acc);
    for (int off = 16; off > 0; off >>= 1) acc += __shfl_xor(acc, off, 32);
    if (lane == 0) att[s] = acc + ca_b[0];
  }
  __syncthreads();

  if (wid == 0) {                                // softmax over 128 (wave 0)
    float mx = -3.4e38f;
    for (int j = 0; j < 4; ++j) mx = fmaxf(mx, att[lane + 32 * j]);
    for (int off = 16; off > 0; off >>= 1) mx = fmaxf(mx, __shfl_xor(mx, off, 32));
    float e[4], ssum = 0.f;
    for (int j = 0; j < 4; ++j) { e[j] = __expf(att[lane + 32 * j] - mx); ssum += e[j]; }
    for (int off = 16; off > 0; off >>= 1) ssum += __shfl_xor(ssum, off, 32);
    const float inv = 1.0f / ssum;
    for (int j = 0; j < 4; ++j) att[lane + 32 * j] = e[j] * inv;
  }
  __syncthreads();

  float hacc = 0.f;
  for (int d = tid; d < 1024; d += 256) {
    float acc = 0.f;
    for (int s = 0; s < 128; ++s) acc = fmaf(att[s], ctx[(size_t)s * 1024 + d], acc);
    control[b * 1024 + d] = acc;
    const float vv = acc * ra_w[d];
    v[b * 1024 + d] = vv;
    hacc = fmaf(vv, ri_b[d], hacc);
  }
  for (int off = 16; off > 0; off >>= 1) hacc += __shfl_xor(hacc, off, 32);
  if (lane == 0) red[wid] = hacc;
  __syncthreads();
  if (tid == 0) {
    float t = 0.f;
    for (int i = 0; i < 8; ++i) t += red[i];
    h[b] = t + ra_b[0];
  }
}

// ---------------------------------------------------------------------------
// Read unit (one block per b). knowledge is [B, D, K] (k contiguous):
//   g[d]    = mem_p[b,d]*w12[b,d] + w12[b,1024+d]
//   att[k]  = softmax_k( sum_d knowledge[b,d,k]*g[d] + h[b] )
//   read[d] = sum_k att[k]*knowledge[b,d,k]
// ---------------------------------------------------------------------------
__global__ __launch_bounds__(256) void read_unit_kernel(
    const float* __restrict__ mem_p, const float* __restrict__ w12,
    const float* __restrict__ knowledge, const float* __restrict__ h,
    float* __restrict__ readout)
{
  __shared__ float g[1024];
  __shared__ float att[512];
  const int b = blockIdx.x, tid = threadIdx.x;
  const int lane = tid & 31, wid = tid >> 5;
  const float* know = knowledge + (size_t)b * 1024 * 512;

  for (int d = tid; d < 1024; d += 256)
    g[d] = mem_p[b * 1024 + d] * w12[b * 2048 + d] + w12[b * 2048 + 1024 + d];
  __syncthreads();

  {                                              // logits; coalesced over k
    float a0 = 0.f, a1 = 0.f;
    for (int d = 0; d < 1024; ++d) {
      const float gd = g[d];
      const float* row = know + (size_t)d * 512;
      a0 = fmaf(gd, row[tid], a0);
      a1 = fmaf(gd, row[tid + 256], a1);
    }
    const float hb = h[b];
    att[tid] = a0 + hb;
    att[tid + 256] = a1 + hb;
  }
  __syncthreads();

  if (wid == 0) {                                // softmax over 512 (wave 0)
    float mx = -3.4e38f;
    for (int i = lane; i < 512; i += 32) mx = fmaxf(mx, att[i]);
    for (int off = 16; off > 0; off >>= 1) mx = fmaxf(mx, __shfl_xor(mx, off, 32));
    float ssum = 0.f;
    for (int i = lane; i < 512; i += 32) { const float e = __expf(att[i] - mx); att[i] = e; ssum += e; }
    for (int off = 16; off > 0; off >>= 1) ssum += __shfl_xor(ssum, off, 32);
    const float inv = 1.0f / ssum;
    for (int i = lane; i < 512; i += 32) att[i] *= inv;
  }
  __syncthreads();

  for (int d = tid; d < 1024; d += 256) {        // read: rows contiguous, b128 loads
    const float4* row = (const float4*)(know + (size_t)d * 512);
    float acc = 0.f;
    for (int k4 = 0; k4 < 128; ++k4) {
      const float4 x = row[k4];
      acc = fmaf(att[4 * k4 + 0], x.x, acc);
      acc = fmaf(att[4 * k4 + 1], x.y, acc);
      acc = fmaf(att[4 * k4 + 2], x.z, acc);
      acc = fmaf(att[4 * k4 + 3], x.w, acc);
    }
    readout[b * 1024 + d] = acc;
  }
}

// ---------------------------------------------------------------------------
extern "C" void kernel_launch(void* const* d_in, const int* in_sizes, int n_in,
                              void* d_out, int out_size, void* d_ws, size_t ws_size,
                              hipStream_t stream) {
  (void)in_sizes; (void)n_in; (void)out_size; (void)ws_size;
  const float* context      = (const float*)d_in[0];   // [32,128,1024]
  const float* question     = (const float*)d_in[1];   // [32,2048]
  const float* knowledge    = (const float*)d_in[2];   // [32,1024,512]
  const float* control_init = (const float*)d_in[3];   // [1,1]
  const float* pa_w = (const float*)d_in[4];           // [12,1024,2048]
  const float* pa_b = (const float*)d_in[5];           // [12,1024]
  const float* cq_w = (const float*)d_in[6];           // [1024,2048]
  const float* cq_b = (const float*)d_in[7];
  const float* ca_w = (const float*)d_in[8];           // [1,1024]
  const float* ca_b = (const float*)d_in[9];
  const float* rm_w = (const float*)d_in[10];          // [1024,1024]
  const float* rm_b = (const float*)d_in[11];
  const float* ri_w = (const float*)d_in[12];          // [1024,2048]
  const float* ri_b = (const float*)d_in[13];
  const float* ra_w = (const float*)d_in[14];          // [1,1024]
  const float* ra_b = (const float*)d_in[15];
  const float* wm_w = (const float*)d_in[16];          // [1024,2048]
  const float* wm_b = (const float*)d_in[17];
  float* out = (float*)d_out;                          // [32,1024] fp32

  float* ws   = (float*)d_ws;                          // ~2.75 MB used
  float* qpa  = ws;                 // 12*32768
  float* ctrl = qpa  + 393216;      // 32768
  float* mem0 = ctrl + 32768;       // 32768
  float* mem1 = mem0 + 32768;       // 32768
  float* cq   = mem1 + 32768;       // 32768
  float* v    = cq   + 32768;       // 32768
  float* memp = v    + 32768;       // 32768
  float* w12  = memp + 32768;       // 65536
  float* hbuf = w12  + 65536;       // 32
  float* rdb  = hbuf + 32;          // 32768

  // q_pa_all[s,b,d] = question[b,:] . pa_w[s,d,:] + pa_b[s,d]
  //   GEMM M=32, N=12288, K=2048 (weights read once, ~100 MB stream)
  wmma_gemm_f32<true><<<dim3(768, 2), 32, 0, stream>>>(
      question, 2048, nullptr, 0, pa_w, 2048, pa_b, qpa, 0, 1);
  init_state_kernel<<<128, 256, 0, stream>>>(control_init, qpa, ctrl, mem0);

  float* mcur = mem0; float* mnxt = mem1;
  for (int s = 0; s < 12; ++s) {
    // cq = concat([control, q_pa_s]) @ cq_w.T + cq_b        (M=32,N=1024,K=2048)
    wmma_gemm_f32<true><<<dim3(64, 2), 32, 0, stream>>>(
        ctrl, 1024, qpa + s * 32768, 1024, cq_w, 2048, cq_b, cq, 1024, 0);
    // context attention -> new control, v, h
    control_unit_kernel<<<32, 256, 0, stream>>>(
        cq, context, ca_w, ca_b, ra_w, ri_b, ra_b, ctrl, v, hbuf);
    // mem_p = memory @ rm_w.T + rm_b                        (M=32,N=1024,K=1024)
    wmma_gemm_f32<true><<<dim3(64, 2), 32, 0, stream>>>(
        mcur, 1024, nullptr, 0, rm_w, 1024, rm_b, memp, 1024, 0);
    // w12 = v @ ri_w                                        (M=32,N=2048,K=1024)
    wmma_gemm_f32<false><<<dim3(128, 2), 32, 0, stream>>>(
        v, 1024, nullptr, 0, ri_w, 2048, nullptr, w12, 2048, 0);
    // knowledge attention -> read
    read_unit_kernel<<<32, 256, 0, stream>>>(memp, w12, knowledge, hbuf, rdb);
    // memory' = concat([read, memory]) @ wm_w.T + wm_b      (M=32,N=1024,K=2048)
    float* dst = (s == 11) ? out : mnxt;
    wmma_gemm_f32<true><<<dim3(64, 2), 32, 0, stream>>>(
        rdb, 1024, mcur, 1024, wm_w, 2048, wm_b, dst, 1024, 0);
    float* t = mcur; mcur = mnxt; mnxt = t;
  }
}